// GraphResNet_85779086836238
// MI455X (gfx1250) — compile-verified
//
#include <hip/hip_runtime.h>
#include <hip/hip_bf16.h>
#include <math.h>

#define NNODES 20000
#define NEDGES 320000
#define NHEADS 4
#define NCHAN  128
#define HC     (NHEADS * NCHAN)   // 512
#define ETOT   (NEDGES + NNODES)  // 340000 (self loops appended)
#define NEG_SLOPE 0.2f

typedef float v2f __attribute__((ext_vector_type(2)));
typedef float v8f __attribute__((ext_vector_type(8)));

// ---------------------------------------------------------------------------
// float atomic max via monotone integer-bit mapping (buffer initialized -inf)
// ---------------------------------------------------------------------------
__device__ __forceinline__ void atomicMaxF(float* addr, float v) {
  if (v >= 0.0f) {
    atomicMax((int*)addr, __float_as_int(v));
  } else {
    atomicMin((unsigned int*)addr, __float_as_uint(v));
  }
}

// ---------------------------------------------------------------------------
// 0) init workspace: m = -inf (first 2*N*H), everything else in range = 0
// ---------------------------------------------------------------------------
__global__ __launch_bounds__(256) void gat_init_kernel(float* ws, int mCount, int total) {
  int i = blockIdx.x * 256 + threadIdx.x;
  if (i >= total) return;
  ws[i] = (i < mCount) ? (-__builtin_inff()) : 0.0f;
}

// ---------------------------------------------------------------------------
// 1) sum(edge_attr) -> eaSum[0]   (wave reduce + 1 atomic per wave)
// ---------------------------------------------------------------------------
__global__ __launch_bounds__(256) void gat_easum_kernel(const float* __restrict__ ea,
                                                        float* __restrict__ eaSum) {
  int i = blockIdx.x * 256 + threadIdx.x;
  float v = (i < NEDGES) ? ea[i] : 0.0f;
  #pragma unroll
  for (int off = 16; off > 0; off >>= 1)
    v += __shfl_xor(v, off, 32);
  if ((threadIdx.x & 31) == 0) atomicAdd(eaSum, v);
}

// ---------------------------------------------------------------------------
// 1b) repack Wfuse [256x128] into K-pair-major pairs: Wp[kp*256 + col*2 + (k&1)]
//     so the fuse kernel's B operand is a single global_load_b64 per WMMA.
// ---------------------------------------------------------------------------
__global__ __launch_bounds__(256) void gat_packwf_kernel(const float* __restrict__ Wf,
                                                         float* __restrict__ Wp) {
  int i = blockIdx.x * 256 + threadIdx.x;   // over 256*128
  if (i >= 256 * NCHAN) return;
  int k = i >> 7;
  int col = i & (NCHAN - 1);
  Wp[(k >> 1) * 256 + col * 2 + (k & 1)] = Wf[k * NCHAN + col];
}

// ---------------------------------------------------------------------------
// 2) pass A: one thread per (edge, head). logit = sum_c att * leakyrelu(t)
//    t = x_s0*Ws0 + x_s1*Ws1 + x_d0*Wd0 + x_d1*Wd1 + ea*We + (bsrc+bdst)
//    then atomic max into m[dst,h].
//    Weights staged interleaved in LDS as 8 floats/channel so the inner loop
//    is 2 x ds_load_b128 instead of 7 x ds_load_b32 (ds-issue bound loop).
// ---------------------------------------------------------------------------
__global__ __launch_bounds__(256) void gat_passA_kernel(
    const float* __restrict__ x, const int* __restrict__ ei,
    const float* __restrict__ ea, const float* __restrict__ eaSum,
    const float* __restrict__ Wsrc, const float* __restrict__ bsrc,
    const float* __restrict__ Wdst, const float* __restrict__ bdst,
    const float* __restrict__ Wedge, const float* __restrict__ att,
    float* __restrict__ logits, float* __restrict__ m) {
  __shared__ float4 sW[HC][2];   // [i][0]={Ws0,Ws1,Wd0,Wd1} [i][1]={We,Bsum,Att,0}
  for (int i = threadIdx.x; i < HC; i += 256) {
    sW[i][0] = make_float4(Wsrc[i], Wsrc[HC + i], Wdst[i], Wdst[HC + i]);
    sW[i][1] = make_float4(Wedge[i], bsrc[i] + bdst[i], att[i], 0.0f);
  }
  __syncthreads();

  int gid = blockIdx.x * 256 + threadIdx.x;
  int e = gid >> 2;
  int h = gid & 3;
  if (e >= ETOT) return;

  int s, d;
  float eav;
  if (e < NEDGES) {
    s = ei[e];
    d = ei[NEDGES + e];
    eav = ea[e];
  } else {               // self loop, fill_value = mean(edge_attr)
    s = d = e - NEDGES;
    eav = eaSum[0] * (1.0f / (float)NEDGES);
  }
  float xs0 = x[2 * s], xs1 = x[2 * s + 1];
  float xd0 = x[2 * d], xd1 = x[2 * d + 1];

  const int base = h * NCHAN;
  float acc = 0.0f;
  #pragma unroll 4
  for (int c = 0; c < NCHAN; ++c) {
    int i = base + c;
    float4 w0 = sW[i][0];
    float4 w1 = sW[i][1];
    float t = fmaf(xs0, w0.x,
              fmaf(xs1, w0.y,
              fmaf(xd0, w0.z,
              fmaf(xd1, w0.w,
              fmaf(eav, w1.x, w1.y)))));
    t = (t > 0.0f) ? t : t * NEG_SLOPE;
    acc = fmaf(w1.z, t, acc);
  }
  logits[(size_t)e * NHEADS + h] = acc;
  atomicMaxF(&m[d * NHEADS + h], acc);
}

// ---------------------------------------------------------------------------
// 3) pass B: p = exp(logit - m[dst]); accumulate rank-2 moments per (dst,h):
//    S0 += x[src,0]*p ; S1 += x[src,1]*p ; Sp += p  (Sp == softmax denom)
// ---------------------------------------------------------------------------
__global__ __launch_bounds__(256) void gat_passB_kernel(
    const float* __restrict__ x, const int* __restrict__ ei,
    const float* __restrict__ logits, const float* __restrict__ m,
    float* __restrict__ S0, float* __restrict__ S1, float* __restrict__ Sp) {
  int gid = blockIdx.x * 256 + threadIdx.x;
  int e = gid >> 2;
  int h = gid & 3;
  if (e >= ETOT) return;

  int s, d;
  if (e < NEDGES) { s = ei[e]; d = ei[NEDGES + e]; }
  else            { s = d = e - NEDGES; }

  float mm = m[d * NHEADS + h];
  if (!(mm > -3.0e38f)) mm = 0.0f;   // reference: non-finite max -> 0
  float p = __expf(logits[(size_t)e * NHEADS + h] - mm);

  float x0 = x[2 * s], x1 = x[2 * s + 1];
  atomicAdd(&S0[d * NHEADS + h], x0 * p);
  atomicAdd(&S1[d * NHEADS + h], x1 * p);
  atomicAdd(&Sp[d * NHEADS + h], p);
}

// ---------------------------------------------------------------------------
// 4) node kernel: h_b[n,c] = relu(mean_h((Ws0*S0 + Ws1*S1 + bs*Sp)/(Sp+eps))
//                 + bias) + x @ Wres ; write into concat buffer Xc[n, b*128+c]
// ---------------------------------------------------------------------------
__global__ __launch_bounds__(256) void gat_node_kernel(
    const float* __restrict__ x,
    const float* __restrict__ S0, const float* __restrict__ S1,
    const float* __restrict__ Sp,
    const float* __restrict__ Wsrc, const float* __restrict__ bsrc,
    const float* __restrict__ bias, const float* __restrict__ Wres,
    float* __restrict__ Xc, int colOff) {
  int gid = blockIdx.x * 256 + threadIdx.x;
  if (gid >= NNODES * NCHAN) return;
  int n = gid >> 7;
  int c = gid & (NCHAN - 1);

  float v = 0.0f;
  #pragma unroll
  for (int h = 0; h < NHEADS; ++h) {
    int i = h * NCHAN + c;
    float s0 = S0[n * NHEADS + h];
    float s1 = S1[n * NHEADS + h];
    float sp = Sp[n * NHEADS + h];
    float inv = 1.0f / (sp + 1e-16f);
    v += (Wsrc[i] * s0 + Wsrc[HC + i] * s1 + bsrc[i] * sp) * inv;
  }
  v = v * (1.0f / (float)NHEADS) + bias[c];
  v = (v > 0.0f) ? v : 0.0f;                                // ReLU
  v += x[2 * n] * Wres[c] + x[2 * n + 1] * Wres[NCHAN + c]; // skip projection
  Xc[(size_t)n * (2 * NCHAN) + colOff + c] = v;
}

// ---------------------------------------------------------------------------
// 5) fuse: relu(Xc[20000x256] @ Wfuse[256x128] + bfuse) with f32 WMMA.
//    One wave per 16x16 output tile, 8 waves/block = one row-tile of tiles.
//    K = 256 -> 64 x v_wmma_f32_16x16x4_f32. A and B are each one b64/step.
// ---------------------------------------------------------------------------
__global__ __launch_bounds__(256) void gat_fuse_wmma_kernel(
    const float* __restrict__ Xc, const float* __restrict__ Wp,
    const float* __restrict__ bfuse, float* __restrict__ out) {
  const int lane = threadIdx.x & 31;
  const int wave = threadIdx.x >> 5;   // col tile 0..7
  const int row0 = blockIdx.x * 16;    // row tile 0..1249
  const int col0 = wave * 16;
  const int lm = lane & 15;            // A row / B,D col within tile
  const int lh = lane >> 4;            // selects K pair (A,B) and row half (D)

  v8f acc = {0.f, 0.f, 0.f, 0.f, 0.f, 0.f, 0.f, 0.f};
  // A lane layout: a[i] = A[lm][k + 2*lh + i]   (16x4 f32 A matrix)
  const float* aBase = Xc + (size_t)(row0 + lm) * 256 + 2 * lh;
  // B lane layout: b[i] = B[k + 2*lh + i][lm]; packed pairs at Wp[kp][col][i]
  const float* bBase = Wp + (size_t)lh * 256 + (size_t)(col0 + lm) * 2;

  #pragma unroll 8
  for (int kk = 0; kk < 64; ++kk) {
    v2f a = *(const v2f*)&aBase[kk * 4];       // global_load_b64
    v2f b = *(const v2f*)&bBase[kk * 512];     // global_load_b64 (kp = kk*2+lh)
    acc = __builtin_amdgcn_wmma_f32_16x16x4_f32(
        /*neg_a=*/false, a, /*neg_b=*/false, b,
        /*c_mod=*/(short)0, acc, /*reuse_a=*/false, /*reuse_b=*/false);
  }

  float bb = bfuse[col0 + lm];
  #pragma unroll
  for (int r = 0; r < 8; ++r) {        // D: VGPR r -> row r + 8*lh, col lm
    float v = acc[r] + bb;
    v = (v > 0.0f) ? v : 0.0f;
    out[(size_t)(row0 + r + 8 * lh) * NCHAN + col0 + lm] = v;
  }
}

// ---------------------------------------------------------------------------
extern "C" void kernel_launch(void* const* d_in, const int* in_sizes, int n_in,
                              void* d_out, int out_size, void* d_ws, size_t ws_size,
                              hipStream_t stream) {
  (void)in_sizes; (void)n_in; (void)out_size; (void)ws_size;
  const float* x  = (const float*)d_in[0];
  const int*   ei = (const int*)d_in[1];
  const float* ea = (const float*)d_in[2];
  const float* Wsrc[2]  = {(const float*)d_in[3],  (const float*)d_in[11]};
  const float* bsrc[2]  = {(const float*)d_in[4],  (const float*)d_in[12]};
  const float* Wdst[2]  = {(const float*)d_in[5],  (const float*)d_in[13]};
  const float* bdst[2]  = {(const float*)d_in[6],  (const float*)d_in[14]};
  const float* Wedge[2] = {(const float*)d_in[7],  (const float*)d_in[15]};
  const float* att[2]   = {(const float*)d_in[8],  (const float*)d_in[16]};
  const float* bias[2]  = {(const float*)d_in[9],  (const float*)d_in[17]};
  const float* Wres[2]  = {(const float*)d_in[10], (const float*)d_in[18]};
  const float* Wfuse = (const float*)d_in[19];
  const float* bfuse = (const float*)d_in[20];
  float* out = (float*)d_out;

  // ---- workspace layout (floats) ----
  const size_t NH = (size_t)NNODES * NHEADS;                 // 80000
  float* ws     = (float*)d_ws;
  float* mArr   = ws;                                        // 2*NH   (-inf)
  float* Sarr   = ws + 2 * NH;                               // 6*NH   (0)
  float* eaSum  = ws + 8 * NH;                               // 16     (0)
  float* logits = ws + 8 * NH + 16;                          // 2*ETOT*NHEADS
  float* Xc     = logits + 2 * (size_t)ETOT * NHEADS;        // NNODES*256
  float* Wpack  = Xc + (size_t)NNODES * 256;                 // 256*128

  const int initTotal = (int)(8 * NH + 16);
  gat_init_kernel<<<(initTotal + 255) / 256, 256, 0, stream>>>(ws, (int)(2 * NH), initTotal);

  gat_easum_kernel<<<(NEDGES + 255) / 256, 256, 0, stream>>>(ea, eaSum);
  gat_packwf_kernel<<<(256 * NCHAN + 255) / 256, 256, 0, stream>>>(Wfuse, Wpack);

  const int edgeThreads = ETOT * NHEADS;                     // 1,360,000
  const int edgeBlocks  = (edgeThreads + 255) / 256;
  const int nodeThreads = NNODES * NCHAN;                    // 2,560,000
  const int nodeBlocks  = (nodeThreads + 255) / 256;

  for (int b = 0; b < 2; ++b) {
    float* mB  = mArr + (size_t)b * NH;
    float* lgB = logits + (size_t)b * ETOT * NHEADS;
    float* S0  = Sarr + (size_t)b * 3 * NH;
    float* S1  = S0 + NH;
    float* Sp  = S1 + NH;

    gat_passA_kernel<<<edgeBlocks, 256, 0, stream>>>(
        x, ei, ea, eaSum, Wsrc[b], bsrc[b], Wdst[b], bdst[b], Wedge[b], att[b],
        lgB, mB);
    gat_passB_kernel<<<edgeBlocks, 256, 0, stream>>>(
        x, ei, lgB, mB, S0, S1, Sp);
    gat_node_kernel<<<nodeBlocks, 256, 0, stream>>>(
        x, S0, S1, Sp, Wsrc[b], bsrc[b], bias[b], Wres[b], Xc, b * NCHAN);
  }

  // fuse GEMM: 1250 row tiles x (8 col tiles == 8 waves per block)
  gat_fuse_wmma_kernel<<<NNODES / 16, 256, 0, stream>>>(Xc, Wpack, bfuse, out);
}